// HGNN_Extractor_49039936585972
// MI455X (gfx1250) — compile-verified
//
#include <hip/hip_runtime.h>

#define HEADS 8
#define EMBED 64
#define PROJ  (HEADS * EMBED)   // 512
#define GAT_EPS 1e-16f

typedef __attribute__((ext_vector_type(2))) float v2f;
typedef __attribute__((ext_vector_type(8))) float v8f;

__device__ __forceinline__ float elu1(float x) {
  return x > 0.f ? x : (__expf(x) - 1.f);
}

// ---------------------------------------------------------------- utilities
__global__ void fill_kernel(float* __restrict__ p, float v, int n) {
  int i = blockIdx.x * blockDim.x + threadIdx.x;
  if (i < n) p[i] = v;
}

// h[n,c] = elu( sum_f x[n,f] * W[c,f] + b[c] )   (F = 4 or 2)
__global__ void embed_kernel(const float* __restrict__ x, const float* __restrict__ W,
                             const float* __restrict__ b, float* __restrict__ h,
                             int n_nodes, int F) {
  int i = blockIdx.x * blockDim.x + threadIdx.x;
  if (i >= n_nodes * EMBED) return;
  int n = i >> 6, c = i & 63;
  float acc = b[c];
  for (int f = 0; f < F; ++f) acc = fmaf(x[n * F + f], W[c * F + f], acc);
  h[i] = elu1(acc);
}

// Fold attention vectors through W:
//   Ai[h,f] = sum_c W[h*64+c, f] * att[h, c]        (dst-side half)
//   Aj[h,f] = sum_c W[h*64+c, f] * att[h, 64+c]     (src-side half)
__global__ void att_fold_kernel(const float* __restrict__ W, const float* __restrict__ att,
                                float* __restrict__ Ai, float* __restrict__ Aj) {
  int t = blockIdx.x * blockDim.x + threadIdx.x;
  if (t >= HEADS * EMBED) return;
  int h = t >> 6, f = t & 63;
  float si = 0.f, sj = 0.f;
  for (int c = 0; c < EMBED; ++c) {
    float w = W[(h * EMBED + c) * EMBED + f];
    si = fmaf(w, att[h * 2 * EMBED + c], si);
    sj = fmaf(w, att[h * 2 * EMBED + EMBED + c], sj);
  }
  Ai[t] = si;
  Aj[t] = sj;
}

// s[n,h] = h[n,:] . A[h,:]
__global__ void score_kernel(const float* __restrict__ hfeat, const float* __restrict__ A,
                             float* __restrict__ s, int n_nodes) {
  int t = blockIdx.x * blockDim.x + threadIdx.x;
  if (t >= n_nodes * HEADS) return;
  int n = t >> 3, h = t & 7;
  const float* hp = hfeat + (size_t)n * EMBED;
  const float* ap = A + h * EMBED;
  float acc = 0.f;
  for (int f = 0; f < EMBED; ++f) acc = fmaf(hp[f], ap[f], acc);
  s[t] = acc;
}

// --------------------------------------------------- fp32 WMMA projection
// hs = h @ W^T : [N, 64] x [64, 512] -> [N, 512]
//
// One wave owns a 16-row strip: A (16x64, the full K) is held in 16 v2f
// registers so h is read from HBM exactly once. W (512x64 = 128 KB) is staged
// into LDS once per workgroup; the inner-loop B feeds are ds_load_b64.
// 32 column tiles x 16 chained V_WMMA_F32_16X16X4_F32 per wave.
//
// A layout (16x4, 32-bit): lanes 0-15 hold M=lane {K0,K1}; lanes 16-31 {K2,K3}.
// B layout (4x16):         lanes 0-15 hold N=lane {K0,K1}; lanes 16-31 {K2,K3}.
__global__ void proj_wmma_kernel(const float* __restrict__ hfeat, const float* __restrict__ W,
                                 float* __restrict__ hs, int n_nodes) {
  extern __shared__ float Ws[];  // PROJ * EMBED floats = 128 KB

  // Cooperative load of W into LDS (b128 per thread), before any wave exits.
  for (int i = threadIdx.x * 4; i < PROJ * EMBED; i += blockDim.x * 4) {
    *(float4*)&Ws[i] = *(const float4*)&W[i];
  }
  __syncthreads();

  int lane = threadIdx.x & 31;
  int rowTile = blockIdx.x * (blockDim.x >> 5) + (threadIdx.x >> 5);
  if (rowTile * 16 >= n_nodes) return;  // wave-uniform: EXEC stays all-ones

  int lo   = lane & 15;
  int koff = (lane >> 4) << 1;  // 0 for lanes 0-15, 2 for lanes 16-31
  int row  = rowTile * 16 + lo;

  // A strip: h[row, 0..63] in the WMMA lane layout, kept for all column tiles.
  v2f a[16];
  const float* ap = hfeat + (size_t)row * EMBED + koff;
#pragma unroll
  for (int kk = 0; kk < 16; ++kk) a[kk] = *(const v2f*)(ap + kk * 4);

  // D layout: VGPR r -> M = r (lanes 0-15) or r+8 (lanes 16-31), N = lane&15
  int mbase = rowTile * 16 + ((lane >> 4) << 3);

  for (int ct = 0; ct < PROJ / 16; ++ct) {
    int cg = ct * 16 + lo;
    const float* bp = Ws + (size_t)cg * EMBED + koff;  // B: W^T[k, cg] = W[cg, k]
    v8f acc = {};
#pragma unroll
    for (int kk = 0; kk < 16; ++kk) {
      v2f b = *(const v2f*)(bp + kk * 4);
      acc = __builtin_amdgcn_wmma_f32_16x16x4_f32(false, a[kk], false, b, (short)0, acc,
                                                  false, false);
    }
#pragma unroll
    for (int r = 0; r < 8; ++r) {
      hs[(size_t)(mbase + r) * PROJ + cg] = acc[r];
    }
  }
}

// ----------------------------------------------------------- edge passes
// Pass 1: alpha[e,h] = s_dst[dst,h] + s_src[src,h]; segment-max into m (float
// atomic max via sign-split int max / uint min).
__global__ void alpha_max_kernel(const int* __restrict__ src, const int* __restrict__ dst,
                                 const float* __restrict__ s_src, const float* __restrict__ s_dst,
                                 float* __restrict__ exw, float* __restrict__ m, int E) {
  int t = blockIdx.x * blockDim.x + threadIdx.x;
  if (t >= E * HEADS) return;
  int e = t >> 3, h = t & 7;
  int d = dst[e];
  float a = s_dst[d * HEADS + h] + s_src[src[e] * HEADS + h];
  exw[t] = a;
  int* ia = (int*)(m + (size_t)d * HEADS + h);
  int av = __float_as_int(a);
  if (av >= 0)
    atomicMax(ia, av);
  else
    atomicMin((unsigned int*)ia, (unsigned int)av);
}

// Pass 2: exw[e,h] = exp(alpha - m[dst,h]); segment-sum into ssum.
__global__ void expsum_kernel(const int* __restrict__ dst, float* __restrict__ exw,
                              const float* __restrict__ m, float* __restrict__ ssum, int E) {
  int t = blockIdx.x * blockDim.x + threadIdx.x;
  if (t >= E * HEADS) return;
  int e = t >> 3, h = t & 7;
  int d = dst[e];
  float v = __expf(exw[t] - m[(size_t)d * HEADS + h]);
  exw[t] = v;
  atomicAdd(&ssum[(size_t)d * HEADS + h], v);
}

// Pass 3: agg[dst,c] += sum_h w[e,h] * hs[src, h*64+c], one atomic per (e,c).
// 4 edges per 256-thread block; w (with 1/HEADS folded in) staged in LDS.
__global__ void scatter_kernel(const int* __restrict__ src, const int* __restrict__ dst,
                               const float* __restrict__ exw, const float* __restrict__ ssum,
                               const float* __restrict__ hs, float* __restrict__ agg, int E) {
  __shared__ float w[4][HEADS];
  int e0 = blockIdx.x * 4;
  int tid = threadIdx.x;
  if (tid < 32) {
    int le = tid >> 3, h = tid & 7;
    int e = e0 + le;
    if (e < E) {
      int d = dst[e];
      w[le][h] = exw[(size_t)e * HEADS + h] /
                 (ssum[(size_t)d * HEADS + h] + GAT_EPS) * (1.0f / HEADS);
    }
  }
  __syncthreads();
  int le = tid >> 6, c = tid & 63;
  int e = e0 + le;
  if (e >= E) return;
  int s = src[e], d = dst[e];
  const float* hp = hs + (size_t)s * PROJ + c;
  float y = 0.f;
#pragma unroll
  for (int h = 0; h < HEADS; ++h) y = fmaf(w[le][h], hp[h * EMBED], y);
  atomicAdd(&agg[(size_t)d * EMBED + c], y);
}

// out = elu(h0 + agg)
__global__ void combine_kernel(const float* __restrict__ h0, const float* __restrict__ agg,
                               float* __restrict__ out, int n) {
  int i = blockIdx.x * blockDim.x + threadIdx.x;
  if (i < n) out[i] = elu1(h0[i] + agg[i]);
}

// ------------------------------------------------------------------ driver
static inline int ceil_div(int a, int b) { return (a + b - 1) / b; }

static void run_gat_layer(hipStream_t stream, int E,
                          const int* src, const int* dst,
                          const float* h_src, int n_src,
                          const float* h_dst0, int n_dst,
                          const float* W, const float* att,
                          float* Ai, float* Aj, float* s_src, float* s_dst,
                          float* m, float* ssum, float* exw, float* hs, float* agg,
                          float* out_dst) {
  const int B = 256;
  att_fold_kernel<<<ceil_div(HEADS * EMBED, B), B, 0, stream>>>(W, att, Ai, Aj);
  score_kernel<<<ceil_div(n_src * HEADS, B), B, 0, stream>>>(h_src, Aj, s_src, n_src);
  score_kernel<<<ceil_div(n_dst * HEADS, B), B, 0, stream>>>(h_dst0, Ai, s_dst, n_dst);

  // 8 waves per block, one 16-row strip per wave; W staged in 128 KB of LDS.
  int rowTiles = ceil_div(n_src, 16);
  size_t lds_bytes = (size_t)PROJ * EMBED * sizeof(float);
  proj_wmma_kernel<<<ceil_div(rowTiles, 8), B, lds_bytes, stream>>>(h_src, W, hs, n_src);

  fill_kernel<<<ceil_div(n_dst * HEADS, B), B, 0, stream>>>(m, -3.0e38f, n_dst * HEADS);
  fill_kernel<<<ceil_div(n_dst * HEADS, B), B, 0, stream>>>(ssum, 0.f, n_dst * HEADS);
  fill_kernel<<<ceil_div(n_dst * EMBED, B), B, 0, stream>>>(agg, 0.f, n_dst * EMBED);

  alpha_max_kernel<<<ceil_div(E * HEADS, B), B, 0, stream>>>(src, dst, s_src, s_dst, exw, m, E);
  expsum_kernel<<<ceil_div(E * HEADS, B), B, 0, stream>>>(dst, exw, m, ssum, E);
  scatter_kernel<<<ceil_div(E, 4), B, 0, stream>>>(src, dst, exw, ssum, hs, agg, E);
  combine_kernel<<<ceil_div(n_dst * EMBED, B), B, 0, stream>>>(h_dst0, agg, out_dst,
                                                               n_dst * EMBED);
}

extern "C" void kernel_launch(void* const* d_in, const int* in_sizes, int n_in,
                              void* d_out, int out_size, void* d_ws, size_t ws_size,
                              hipStream_t stream) {
  const float* op_nodes = (const float*)d_in[0];
  const float* mc_nodes = (const float*)d_in[1];
  const int*   eidx     = (const int*)d_in[2];
  const float* W_op     = (const float*)d_in[3];
  const float* b_op     = (const float*)d_in[4];
  const float* W_mc     = (const float*)d_in[5];
  const float* b_mc     = (const float*)d_in[6];
  const float* W_om     = (const float*)d_in[7];
  const float* att_om   = (const float*)d_in[8];
  const float* W_mo     = (const float*)d_in[9];
  const float* att_mo   = (const float*)d_in[10];

  int n_op = in_sizes[0] / 4;
  int n_mc = in_sizes[1] / 2;
  int E    = in_sizes[2] / 2;
  int n_max = n_op > n_mc ? n_op : n_mc;

  // ---- workspace layout (floats) ----
  float* ws = (float*)d_ws;
  size_t off = 0;
  float* h_op0 = ws + off; off += (size_t)n_op * EMBED;
  float* h_mc0 = ws + off; off += (size_t)n_mc * EMBED;
  float* hs    = ws + off; off += (size_t)n_max * PROJ;   // projection buffer (reused)
  float* s_src = ws + off; off += (size_t)n_max * HEADS;
  float* s_dst = ws + off; off += (size_t)n_max * HEADS;
  float* m     = ws + off; off += (size_t)n_max * HEADS;
  float* ssum  = ws + off; off += (size_t)n_max * HEADS;
  float* exw   = ws + off; off += (size_t)E * HEADS;
  float* agg   = ws + off; off += (size_t)n_max * EMBED;
  float* Ai    = ws + off; off += HEADS * EMBED;
  float* Aj    = ws + off; off += HEADS * EMBED;
  if (ws_size < off * sizeof(float)) return;  // insufficient scratch: bail deterministically

  float* out_op = (float*)d_out;                         // [n_op, 64]
  float* out_mc = (float*)d_out + (size_t)n_op * EMBED;  // [n_mc, 64]

  const int B = 256;
  // Input embeddings
  embed_kernel<<<ceil_div(n_op * EMBED, B), B, 0, stream>>>(op_nodes, W_op, b_op, h_op0,
                                                            n_op, 4);
  embed_kernel<<<ceil_div(n_mc * EMBED, B), B, 0, stream>>>(mc_nodes, W_mc, b_mc, h_mc0,
                                                            n_mc, 2);

  const int* src_om = eidx;      // op ids
  const int* dst_om = eidx + E;  // mc ids

  // Layer 1: op -> mc; h_mc1 = elu(h_mc0 + gat) written straight to d_out mc region
  run_gat_layer(stream, E, src_om, dst_om, h_op0, n_op, h_mc0, n_mc, W_om, att_om,
                Ai, Aj, s_src, s_dst, m, ssum, exw, hs, agg, out_mc);

  // Layer 2: mc -> op (edges reversed); source features are the updated h_mc1
  run_gat_layer(stream, E, dst_om, src_om, out_mc, n_mc, h_op0, n_op, W_mo, att_mo,
                Ai, Aj, s_src, s_dst, m, ssum, exw, hs, agg, out_op);
}